// QuadClassifier0_22411139350996
// MI455X (gfx1250) — compile-verified
//
#include <hip/hip_runtime.h>

#define S 2048
#define NBATCH 32
#define CHUNKS 16          // blocks per (batch, quadrant)
#define NBLOCKS (NBATCH * 4 * CHUNKS)   // 2048
#define ROWS_PER_BLOCK (1024 / CHUNKS)  // 64 half-rows of 1024 floats

typedef float v2f __attribute__((ext_vector_type(2)));
typedef float v4f __attribute__((ext_vector_type(4)));
typedef float v8f __attribute__((ext_vector_type(8)));

// ---------------------------------------------------------------------------
// Pass 1: streaming quadrant reduction. Each block owns 64 contiguous
// half-rows (1024 floats each) of one (batch, quadrant); writes ONE partial.
// Deterministic: fixed per-thread serial order, fixed shuffle tree, fixed
// LDS-reduce order, unique output slot per block (no atomics).
// ---------------------------------------------------------------------------
__global__ __launch_bounds__(256) void quad_reduce_kernel(
    const float* __restrict__ x, float* __restrict__ partial)
{
    const int b     = blockIdx.x;        // 0..2047
    const int n     = b >> 6;            // batch
    const int rem   = b & 63;
    const int q     = rem >> 4;          // quadrant 0..3
    const int chunk = rem & 15;

    // quadrant geometry matches reference quadrant_ids():
    // q0: r<1024,c<1024  q1: r>=1024,c<1024  q2: r>=1024,c>=1024  q3: r<1024,c>=1024
    const int rowBase = (q == 1 || q == 2) ? 1024 : 0;
    const int colBase = (q >= 2) ? 1024 : 0;

    const float* base = x + (size_t)n * S * S
                          + (size_t)(rowBase + chunk * ROWS_PER_BLOCK) * S
                          + colBase + threadIdx.x * 4;

    float acc = 0.0f;
#pragma unroll 8
    for (int r = 0; r < ROWS_PER_BLOCK; ++r) {
        v4f val = __builtin_nontemporal_load((const v4f*)(base + (size_t)r * S));
        acc += val.x; acc += val.y; acc += val.z; acc += val.w;
    }

    // wave32 butterfly reduction
#pragma unroll
    for (int off = 16; off > 0; off >>= 1)
        acc += __shfl_xor(acc, off, 32);

    __shared__ float wsum[8];
    const int wave = threadIdx.x >> 5;
    const int lane = threadIdx.x & 31;
    if (lane == 0) wsum[wave] = acc;
    __syncthreads();
    if (threadIdx.x == 0) {
        float s = 0.0f;
#pragma unroll
        for (int w = 0; w < 8; ++w) s += wsum[w];
        partial[b] = s;  // layout: [(n*4+q)*CHUNKS + chunk]
    }
}

// ---------------------------------------------------------------------------
// Pass 2: one wave. Fold partials -> sums(32,4); weight-norm linear ->
// h(32,16) in LDS; then 32x16 @ 16x10 FC via V_WMMA_F32_16X16X4_F32
// (two 16x16 M-tiles, K=16 in four x4 steps), add b_fc, store (32,10).
// ---------------------------------------------------------------------------
__global__ __launch_bounds__(32) void quad_finish_kernel(
    const float* __restrict__ partial,
    const float* __restrict__ v,      // (4,4,1)
    const float* __restrict__ g,      // (4)
    const float* __restrict__ b_fgl,  // (4,4)
    const float* __restrict__ Wfc,    // (10,16)
    const float* __restrict__ bfc,    // (10)
    float* __restrict__ out)          // (32,10)
{
    __shared__ float hbuf[NBATCH][16];
    const int l = threadIdx.x;        // lane == batch for stage 1

    // per-quadrant sums for batch l, fixed summation order
    float sums[4];
#pragma unroll
    for (int q = 0; q < 4; ++q) {
        float s = 0.0f;
        for (int c = 0; c < CHUNKS; ++c) s += partial[(l * 4 + q) * CHUNKS + c];
        sums[q] = s;
    }
    // weight-normed per-node linear: h[n, o*4+c] = sums[n,o]*g[o]*v[o,c]/||v[o]|| + b[o,c]
#pragma unroll
    for (int o = 0; o < 4; ++o) {
        float n2 = 0.0f;
#pragma unroll
        for (int c = 0; c < 4; ++c) { float vv = v[o * 4 + c]; n2 += vv * vv; }
        const float scale = g[o] / sqrtf(n2);
#pragma unroll
        for (int c = 0; c < 4; ++c)
            hbuf[l][o * 4 + c] = sums[o] * (v[o * 4 + c] * scale) + b_fgl[o * 4 + c];
    }
    __syncthreads();

    // WMMA fragment indices (ISA 7.12.2, 32-bit 16x4 A / 4x16 B / 16x16 C-D):
    //   A[M,K]: lane = M + 16*(K>=2), vgpr = K%2  -> K = k0 + 2*half + vgpr
    //   B[K,N]: lane = N + 16*(K>=2), vgpr = K%2
    //   D[M,N]: lane = N + 16*mhalf,  vgpr vv -> M = vv + 8*mhalf
    const int nIdx = l & 15;   // N (class column) / M row within tile
    const int half = l >> 4;

    // B = Wfc^T padded to 16 columns: B[k, cls] = Wfc[cls, k], cls>=10 -> 0
    v2f Bfr[4];
#pragma unroll
    for (int kb = 0; kb < 4; ++kb) {
        const int kA = kb * 4 + 2 * half;
        Bfr[kb].x = (nIdx < 10) ? Wfc[nIdx * 16 + kA]     : 0.0f;
        Bfr[kb].y = (nIdx < 10) ? Wfc[nIdx * 16 + kA + 1] : 0.0f;
    }
    const float bias = (nIdx < 10) ? bfc[nIdx] : 0.0f;

#pragma unroll
    for (int t = 0; t < 2; ++t) {       // M-tiles: batches 0-15, 16-31
        v8f C = {};
#pragma unroll
        for (int kb = 0; kb < 4; ++kb) {
            v2f A;
            const int row = t * 16 + nIdx;          // A row M = lane%16
            const int kA  = kb * 4 + 2 * half;
            A.x = hbuf[row][kA];
            A.y = hbuf[row][kA + 1];
            C = __builtin_amdgcn_wmma_f32_16x16x4_f32(
                    false, A, false, Bfr[kb], (short)0, C, false, false);
        }
        if (nIdx < 10) {
#pragma unroll
            for (int vv = 0; vv < 8; ++vv) {
                const int M = vv + 8 * half;
                out[(t * 16 + M) * 10 + nIdx] = C[vv] + bias;
            }
        }
    }
}

// ---------------------------------------------------------------------------
extern "C" void kernel_launch(void* const* d_in, const int* in_sizes, int n_in,
                              void* d_out, int out_size, void* d_ws, size_t ws_size,
                              hipStream_t stream) {
    const float* x     = (const float*)d_in[0];  // (32, 2048, 2048)
    const float* v     = (const float*)d_in[1];  // (4,4,1)
    const float* g     = (const float*)d_in[2];  // (4,)
    const float* b_fgl = (const float*)d_in[3];  // (4,4)
    const float* Wfc   = (const float*)d_in[4];  // (10,16)
    const float* bfc   = (const float*)d_in[5];  // (10,)
    // d_in[6] = quads: geometry is hardcoded to match quadrant_ids()

    float* partial = (float*)d_ws;               // NBLOCKS floats (8 KB)

    quad_reduce_kernel<<<NBLOCKS, 256, 0, stream>>>(x, partial);
    quad_finish_kernel<<<1, 32, 0, stream>>>(partial, v, g, b_fgl, Wfc, bfc,
                                             (float*)d_out);
}